// LocalAttnDecoderRNN_38620345926141
// MI455X (gfx1250) — compile-verified
//
#include <hip/hip_runtime.h>
#include <math.h>

// ---------------------------------------------------------------------------
// LocalAttnDecoderRNN single step, MI455X (gfx1250, wave32).
//
// Roofline reasoning:
//  * log_softmax over a size-1 axis is identically zero => the 50257x1024
//    output projection (206 MB, ~80% of reference traffic) is dead. Skipped.
//  * Live traffic: attn_W 4MB + enc 2MB + comb_W 8MB + W_ih/W_hh 32MB ~ 46MB
//    fp32 streamed once => ~2us at 23.3 TB/s. Bandwidth bound, fp32 kept;
//    default (RT) temporal hints keep the 46MB working set resident in the
//    192MB L2 across graph replays.
//  * All GEMMs are matvecs; padded into V_WMMA_F32_16X16X4_F32 tiles
//    (16 outputs per wave, K stepped by 4). Vector operand is built
//    branchlessly (uniform load + v_cndmask select) so the inner loop has no
//    EXEC-mask save/restore branches around the WMMA.
// ---------------------------------------------------------------------------

typedef __attribute__((ext_vector_type(2))) float v2f;
typedef __attribute__((ext_vector_type(8))) float v8f;

#define HDIM   1024
#define VOCAB  50257
#define LDIM   512
#define WWIDTH 5.0f

// workspace layout (float offsets), total 10240 floats = 40 KB
#define EMB0_OFF  0      // emb (1024)                    -> cat1 = [emb | h]
#define HVEC_OFF  1024   // h   (1024)
#define EMB1_OFF  2048   // emb copy (1024)               -> cat2 = [emb | attn_applied]
#define APP_OFF   3072   // attn_applied (1024)
#define AW_OFF    4096   // aw (512)
#define LOC_OFF   4608   // local (512)
#define X_OFF     5120   // x (1024)
#define GATE_OFF  6144   // gates (4096)

// ---------------- logits are exactly zero (log_softmax of singleton) -------
__global__ void zero_logits_k(float* __restrict__ out) {
    int i = blockIdx.x * blockDim.x + threadIdx.x;
    if (i < VOCAB) out[i] = 0.0f;
}

// ---------------- gather embedding row, stage vectors into ws --------------
__global__ void prep_k(const int* __restrict__ token,
                       const float* __restrict__ emb_W,
                       const float* __restrict__ h0,
                       float* __restrict__ ws) {
    int i = blockIdx.x * blockDim.x + threadIdx.x;
    if (i < HDIM) {
        float e = emb_W[(size_t)token[0] * HDIM + i];
        ws[EMB0_OFF + i] = e;
        ws[EMB1_OFF + i] = e;
        ws[HVEC_OFF + i] = h0[i];
    }
}

// ---------------- WMMA matvec core:  out[row0..row0+15] += W(16xK) . v -----
// A = weight tile (16x4 fp32), B = vector in column 0, C accumulates.
// A layout: lane m (0..15) VGPR{0,1} = K{0,1}; lane 16+m = K{2,3}.
// B layout: lane 0 = {v[k],v[k+1]}, lane 16 = {v[k+2],v[k+3]}, others 0.
// All lanes load the (half-uniform) vector chunk; lanes m!=0 select it to
// zero via v_cndmask -- no divergent branch in the inner loop.
__device__ __forceinline__ v8f wmma_rows_acc(const float* __restrict__ W, int ldw,
                                             const float* __restrict__ v, int K,
                                             int row0, v8f c) {
    const int  lane = threadIdx.x & 31;
    const int  m    = lane & 15;
    const int  half = lane >> 4;
    const bool lead = (m == 0);
    const float* wrow = W + (size_t)(row0 + m) * ldw;
    for (int k = 0; k < K; k += 4) {
        v2f a  = *(const v2f*)(wrow + k + 2 * half);
        v2f bv = *(const v2f*)(v    + k + 2 * half);   // broadcast within half
        v2f b;
        b.x = lead ? bv.x : 0.0f;
        b.y = lead ? bv.y : 0.0f;
        c = __builtin_amdgcn_wmma_f32_16x16x4_f32(false, a, false, b,
                                                  (short)0, c, false, false);
    }
    return c;
}

// D column 0 of the 16x16 tile: lane 0 holds M=0..7 in c[0..7], lane 16 M=8..15.
__device__ __forceinline__ void wmma_rows_store(float* __restrict__ out,
                                                const float* __restrict__ bias,
                                                int row0, v8f c) {
    const int lane = threadIdx.x & 31;
    if ((lane & 15) == 0) {
        int base = row0 + (lane >> 4) * 8;
#pragma unroll
        for (int r = 0; r < 8; ++r)
            out[base + r] = c[r] + (bias ? bias[base + r] : 0.0f);
    }
}

// out = W(MxK) . v + bias ; one wave per 16 rows, blockDim=256 (8 waves).
__global__ void matvec_rows_k(const float* __restrict__ W,
                              const float* __restrict__ v,
                              const float* __restrict__ bias,
                              float* __restrict__ out, int K) {
    int wave = (blockIdx.x * blockDim.x + threadIdx.x) >> 5;
    int row0 = wave * 16;
    v8f c = {};
    c = wmma_rows_acc(W, K, v, K, row0, c);
    wmma_rows_store(out, bias, row0, c);
}

// gates = W_ih.x + W_hh.h + b_ih + b_hh, fused accumulation in one C tile.
__global__ void lstm_gates_k(const float* __restrict__ W_ih,
                             const float* __restrict__ W_hh,
                             const float* __restrict__ x,
                             const float* __restrict__ h,
                             const float* __restrict__ b_ih,
                             const float* __restrict__ b_hh,
                             float* __restrict__ gates) {
    int wave = (blockIdx.x * blockDim.x + threadIdx.x) >> 5;
    int row0 = wave * 16;
    v8f c = {};
    c = wmma_rows_acc(W_ih, HDIM, x, HDIM, row0, c);
    c = wmma_rows_acc(W_hh, HDIM, h, HDIM, row0, c);
    const int lane = threadIdx.x & 31;
    if ((lane & 15) == 0) {
        int base = row0 + (lane >> 4) * 8;
#pragma unroll
        for (int r = 0; r < 8; ++r)
            gates[base + r] = c[r] + b_ih[base + r] + b_hh[base + r];
    }
}

// out[j0..j0+15] = v(1xK) . Wc(KxN) ; vector in row 0 of A, Wc tile in B.
// B layout: lane n loads rows k+2*half, k+2*half+1 at column j0+n (coalesced
// across the 16 lanes of each half-wave). A operand built branchlessly.
__global__ void matvec_cols_k(const float* __restrict__ Wc, // K x N row-major
                              const float* __restrict__ v,
                              float* __restrict__ out, int K, int N) {
    int wave = (blockIdx.x * blockDim.x + threadIdx.x) >> 5;
    int j0 = wave * 16;
    const int  lane = threadIdx.x & 31;
    const int  n    = lane & 15;
    const int  half = lane >> 4;
    const bool lead = (n == 0);
    v8f c = {};
    for (int k = 0; k < K; k += 4) {
        v2f av = *(const v2f*)(v + k + 2 * half);      // broadcast within half
        v2f a;
        a.x = lead ? av.x : 0.0f;
        a.y = lead ? av.y : 0.0f;
        v2f b;
        b.x = Wc[(size_t)(k + 2 * half)     * N + j0 + n];
        b.y = Wc[(size_t)(k + 2 * half + 1) * N + j0 + n];
        c = __builtin_amdgcn_wmma_f32_16x16x4_f32(false, a, false, b,
                                                  (short)0, c, false, false);
    }
    if (half == 0) out[j0 + n] = c[0];   // D row M=0: VGPR0, lanes 0..15
}

// center = aw . attn_lin_W + b ; local = aw * triangular window(center).
__global__ void localize_k(const float* __restrict__ aw,
                           const float* __restrict__ lin_W,
                           const float* __restrict__ lin_b,
                           float* __restrict__ local) {
    __shared__ float red[LDIM];
    int t = threadIdx.x;                       // blockDim = 512, one block
    red[t] = aw[t] * lin_W[t];
    __syncthreads();
    for (int s = LDIM / 2; s > 0; s >>= 1) {
        if (t < s) red[t] += red[t + s];
        __syncthreads();
    }
    float center = red[0] + lin_b[0];
    float p  = (float)t;
    float wv = fmaxf(p - center + WWIDTH, 0.0f) *
               fmaxf(-p + center + WWIDTH, 0.0f) * (1.0f / (WWIDTH * WWIDTH));
    local[t] = aw[t] * wv;
}

// PyTorch LSTM step, gate order i,f,g,o; writes h_new, c_new into d_out tail.
__global__ void lstm_pointwise_k(const float* __restrict__ gates,
                                 const float* __restrict__ c0,
                                 float* __restrict__ out) {
    int j = blockIdx.x * blockDim.x + threadIdx.x;
    if (j < HDIM) {
        float gi = gates[j];
        float gf = gates[HDIM + j];
        float gg = gates[2 * HDIM + j];
        float go = gates[3 * HDIM + j];
        float si = 1.0f / (1.0f + expf(-gi));
        float sf = 1.0f / (1.0f + expf(-gf));
        float so = 1.0f / (1.0f + expf(-go));
        float cn = sf * c0[j] + si * tanhf(gg);
        float hn = so * tanhf(cn);
        out[VOCAB + j]        = hn;   // h_new
        out[VOCAB + HDIM + j] = cn;   // c_new
    }
}

extern "C" void kernel_launch(void* const* d_in, const int* in_sizes, int n_in,
                              void* d_out, int out_size, void* d_ws, size_t ws_size,
                              hipStream_t stream) {
    (void)in_sizes; (void)n_in; (void)out_size; (void)ws_size;
    const int*   token  = (const int*)  d_in[0];
    const float* h0     = (const float*)d_in[1];
    const float* c0     = (const float*)d_in[2];
    const float* enc    = (const float*)d_in[3];   // (L, H)
    const float* emb_W  = (const float*)d_in[4];   // (V, H)
    const float* attn_W = (const float*)d_in[5];   // (L, 2H)
    const float* attn_b = (const float*)d_in[6];   // (L,)
    const float* lin_W  = (const float*)d_in[7];   // (1, L)
    const float* lin_b  = (const float*)d_in[8];   // (1,)
    const float* comb_W = (const float*)d_in[9];   // (H, 2H)
    const float* comb_b = (const float*)d_in[10];  // (H,)
    const float* W_ih   = (const float*)d_in[11];  // (4H, H)
    const float* W_hh   = (const float*)d_in[12];  // (4H, H)
    const float* b_ih   = (const float*)d_in[13];  // (4H,)
    const float* b_hh   = (const float*)d_in[14];  // (4H,)
    // d_in[15]=out_W, d_in[16]=out_b intentionally unused:
    // log_softmax over a singleton axis makes the logits output exactly 0.

    float* out = (float*)d_out;
    float* ws  = (float*)d_ws;

    // logp = 0 (dead output projection)
    zero_logits_k<<<(VOCAB + 255) / 256, 256, 0, stream>>>(out);

    // stage emb (twice, for both concatenations) and h
    prep_k<<<(HDIM + 255) / 256, 256, 0, stream>>>(token, emb_W, h0, ws);

    // aw = attn_W (512x2048) . [emb|h] + attn_b        : 512/128 = 4 blocks
    matvec_rows_k<<<4, 256, 0, stream>>>(attn_W, ws + EMB0_OFF, attn_b,
                                         ws + AW_OFF, 2 * HDIM);

    // center + triangular window -> local
    localize_k<<<1, LDIM, 0, stream>>>(ws + AW_OFF, lin_W, lin_b, ws + LOC_OFF);

    // attn_applied = local^T . enc (512x1024)          : 1024/16/8 = 8 blocks
    matvec_cols_k<<<8, 256, 0, stream>>>(enc, ws + LOC_OFF, ws + APP_OFF,
                                         LDIM, HDIM);

    // x = comb_W (1024x2048) . [emb|attn_applied] + b  : 1024/128 = 8 blocks
    matvec_rows_k<<<8, 256, 0, stream>>>(comb_W, ws + EMB1_OFF, comb_b,
                                         ws + X_OFF, 2 * HDIM);

    // gates = W_ih.x + W_hh.h + biases                 : 4096/128 = 32 blocks
    lstm_gates_k<<<32, 256, 0, stream>>>(W_ih, W_hh, ws + X_OFF, ws + HVEC_OFF,
                                         b_ih, b_hh, ws + GATE_OFF);

    // pointwise LSTM -> h_new, c_new
    lstm_pointwise_k<<<(HDIM + 255) / 256, 256, 0, stream>>>(ws + GATE_OFF, c0, out);
}